// LDRSubdiagonal_10419590660370
// MI455X (gfx1250) — compile-verified
//
#include <hip/hip_runtime.h>

// LDR subdiagonal structured matmul for MI455X (gfx1250, wave32, WMMA).
// out = x @ M^T + bias, M = sum_r K(A,g_r) K(B,h_r)^T, K = scaled lower-tri Toeplitz.
// Pass1: tmp[r,j,b] = sum_{k>=j} (H[r,k-j]/PB[k-j]) * (PB[k]*x[b,k])
// Pass2: out[b,i]   = bias[i] + PA[i] * sum_r sum_{j<=i} (G[r,i-j]/PA[i-j]) * tmp[r,j,b]
// f32 WMMA 16x16x4; Toeplitz operand generated from a zero-padded LDS vector.
// Register blocking: 8 accumulators per wave (2 Toeplitz sub-tiles x 4 batch sub-tiles)
// -> 1 global b32 load per v_wmma (streamed operand reused across the 2 sub-tiles).

typedef __attribute__((ext_vector_type(2))) float v2f;
typedef __attribute__((ext_vector_type(8))) float v8f;

#define PAD 32  // front zero padding of the LDS Toeplitz vector (covers d >= -31)

// ---------------- prefix products PA[i] = prod_{t<i} subdA[t] ----------------
__global__ void k_prefix(const float* __restrict__ sA, const float* __restrict__ sB,
                         float* __restrict__ PA, float* __restrict__ PB, int n) {
  if (threadIdx.x == 0) {
    float p = 1.f; PA[0] = p;
    for (int i = 1; i < n; ++i) { p *= sA[i - 1]; PA[i] = p; }
  } else if (threadIdx.x == 1) {
    float p = 1.f; PB[0] = p;
    for (int i = 1; i < n; ++i) { p *= sB[i - 1]; PB[i] = p; }
  }
}

// ---------------- Xt[k*batch + b] = PB[k] * x[b*n + k] (tiled transpose) -----
__global__ void k_transpose_scale(const float* __restrict__ x, const float* __restrict__ PB,
                                  float* __restrict__ Xt, int n, int batch) {
  __shared__ float tile[32][33];
  const int k0 = blockIdx.x * 32, b0 = blockIdx.y * 32;
  const int tx = threadIdx.x, ty = threadIdx.y;  // block (32, 8)
#pragma unroll
  for (int s = 0; s < 32; s += 8)
    tile[ty + s][tx] = x[(size_t)(b0 + ty + s) * n + (k0 + tx)];
  __syncthreads();
#pragma unroll
  for (int s = 0; s < 32; s += 8) {
    const int k = k0 + ty + s;
    Xt[(size_t)k * batch + (b0 + tx)] = PB[k] * tile[tx][ty + s];
  }
}

// ---------------- Pass 1: tmp[r,j,b] -----------------------------------------
// grid (n/64, batch/256, R), block 256 (8 waves).
// Wave w: j-sub (w&1)*32 (two 16-row WMMA tiles), b-sub (w>>1)*64 (four 16-col tiles).
__global__ void k_pass1(const float* __restrict__ Hm, const float* __restrict__ PB,
                        const float* __restrict__ Xt, float* __restrict__ TMP,
                        int n, int batch) {
  extern __shared__ float hp[];  // PAD + n + 4 floats, hp[PAD + m] = H[r,m]/PB[m]
  const int tid = threadIdx.x;
  const int lane = tid & 31;
  const int w = tid >> 5;
  const int r = blockIdx.z;

  for (int t = tid; t < PAD; t += blockDim.x) hp[t] = 0.f;
  for (int t = tid; t < 4; t += blockDim.x) hp[PAD + n + t] = 0.f;
  for (int t = tid; t < n; t += blockDim.x) hp[PAD + t] = Hm[(size_t)r * n + t] / PB[t];
  __syncthreads();

  const int j0 = blockIdx.x * 64 + (w & 1) * 32;   // wave's 32-row j block
  const int b0 = blockIdx.y * 256 + (w >> 1) * 64; // wave's 64-col b block

  const int jr   = lane & 15;        // A row  (M = j)
  const int kb2  = (lane >> 4) * 2;  // K half-wave offset (K = kb2, kb2+1)
  const int nlan = lane & 15;        // B col  (N = b)

  v8f c[2][4];
#pragma unroll
  for (int jt = 0; jt < 2; ++jt)
#pragma unroll
    for (int t = 0; t < 4; ++t) c[jt][t] = (v8f){0.f,0.f,0.f,0.f,0.f,0.f,0.f,0.f};

  // hp index for A sub-tile jt at k-step k0: abase[jt] + k0 (A.x), +1 (A.y)
  const int abase0 = PAD + kb2 - jr - j0;
  const int abase1 = abase0 - 16;
  const float* __restrict__ xcol = Xt + (size_t)kb2 * batch + (b0 + nlan);

  for (int k0 = j0; k0 < n; k0 += 4) {   // triangular: k >= wave's j block start
    v2f a0, a1;
    a0.x = hp[abase0 + k0]; a0.y = hp[abase0 + k0 + 1];
    a1.x = hp[abase1 + k0]; a1.y = hp[abase1 + k0 + 1];
    const float* __restrict__ xp = xcol + (size_t)k0 * batch;
    __builtin_prefetch(xp + (size_t)32 * batch, 0, 1);  // global_prefetch_b8
#pragma unroll
    for (int t = 0; t < 4; ++t) {
      v2f bb;
      bb.x = xp[16 * t];
      bb.y = xp[batch + 16 * t];
      c[0][t] = __builtin_amdgcn_wmma_f32_16x16x4_f32(false, a0, false, bb,
                                                      (short)0, c[0][t], false, false);
      c[1][t] = __builtin_amdgcn_wmma_f32_16x16x4_f32(false, a1, false, bb,
                                                      (short)0, c[1][t], false, false);
    }
  }

  // C layout: VGPR v, lanes 0-15 -> (M=v, N=lane), lanes 16-31 -> (M=v+8, N=lane-16)
  const int jhalf = (lane >> 4) * 8;
#pragma unroll
  for (int jt = 0; jt < 2; ++jt)
#pragma unroll
    for (int t = 0; t < 4; ++t)
#pragma unroll
      for (int v = 0; v < 8; ++v) {
        const int j = j0 + 16 * jt + v + jhalf;
        const int b = b0 + nlan + 16 * t;
        TMP[((size_t)r * n + j) * batch + b] = c[jt][t][v];
      }
}

// ---------------- Pass 2: out[b,i] = bias + PA_i * sum_r (Gp * tmp) ----------
// grid (n/64, batch/256), block 256 (8 waves).
// Wave w: i-sub (w&1)*32 (two 16-col WMMA tiles), b-sub (w>>1)*64 (four 16-row tiles).
// A = tmp^T (M=b, K=j) from global (reused across the 2 i tiles); B = Toeplitz(G') from LDS.
__global__ void k_pass2(const float* __restrict__ Gm, const float* __restrict__ PA,
                        const float* __restrict__ TMP, const float* __restrict__ bias,
                        float* __restrict__ out, int n, int batch, int rdim) {
  extern __shared__ float gp[];  // PAD + n + 4
  const int tid = threadIdx.x;
  const int lane = tid & 31;
  const int w = tid >> 5;

  const int i0 = blockIdx.x * 64 + (w & 1) * 32;
  const int b0 = blockIdx.y * 256 + (w >> 1) * 64;

  const int mlan = lane & 15;        // A row (M = b) and B col (N = i)
  const int kb2  = (lane >> 4) * 2;  // K = j offset

  v8f c[2][4];
#pragma unroll
  for (int it = 0; it < 2; ++it)
#pragma unroll
    for (int t = 0; t < 4; ++t) c[it][t] = (v8f){0.f,0.f,0.f,0.f,0.f,0.f,0.f,0.f};

  const int jmax   = i0 + 32;                 // covers j <= i for both i sub-tiles (pad masks rest)
  const int bbase0 = PAD + i0 + mlan - kb2;   // gp index = bbase - j0k (B.x), -1 (B.y)
  const int bbase1 = bbase0 + 16;

  for (int r = 0; r < rdim; ++r) {
    __syncthreads();
    for (int t = tid; t < PAD; t += blockDim.x) gp[t] = 0.f;
    for (int t = tid; t < 4; t += blockDim.x) gp[PAD + n + t] = 0.f;
    for (int t = tid; t < n; t += blockDim.x) gp[PAD + t] = Gm[(size_t)r * n + t] / PA[t];
    __syncthreads();

    const float* __restrict__ trow =
        TMP + ((size_t)r * n + kb2) * batch + (b0 + mlan);
    for (int j0k = 0; j0k < jmax; j0k += 4) {
      v2f bb0, bb1;
      bb0.x = gp[bbase0 - j0k]; bb0.y = gp[bbase0 - j0k - 1];
      bb1.x = gp[bbase1 - j0k]; bb1.y = gp[bbase1 - j0k - 1];
      const float* __restrict__ ap = trow + (size_t)j0k * batch;
#pragma unroll
      for (int t = 0; t < 4; ++t) {
        v2f a;
        a.x = ap[16 * t];
        a.y = ap[batch + 16 * t];
        c[0][t] = __builtin_amdgcn_wmma_f32_16x16x4_f32(false, a, false, bb0,
                                                        (short)0, c[0][t], false, false);
        c[1][t] = __builtin_amdgcn_wmma_f32_16x16x4_f32(false, a, false, bb1,
                                                        (short)0, c[1][t], false, false);
      }
    }
  }

  const int bhalf = (lane >> 4) * 8;
#pragma unroll
  for (int it = 0; it < 2; ++it) {
    const int i = i0 + 16 * it + mlan;
    const float pa = PA[i];
    const float bi = bias[i];
#pragma unroll
    for (int t = 0; t < 4; ++t)
#pragma unroll
      for (int v = 0; v < 8; ++v) {
        const int b = b0 + 16 * t + v + bhalf;
        out[(size_t)b * n + i] = bi + pa * c[it][t][v];
      }
  }
}

// ---------------- launcher ---------------------------------------------------
extern "C" void kernel_launch(void* const* d_in, const int* in_sizes, int n_in,
                              void* d_out, int out_size, void* d_ws, size_t ws_size,
                              hipStream_t stream) {
  const float* x     = (const float*)d_in[0];
  const float* sA    = (const float*)d_in[1];
  const float* sB    = (const float*)d_in[2];
  const float* G     = (const float*)d_in[3];
  const float* H     = (const float*)d_in[4];
  const float* bias  = (const float*)d_in[5];
  float* out = (float*)d_out;

  const int n     = in_sizes[5];            // 4096
  const int batch = in_sizes[0] / n;        // 1024
  const int rdim  = in_sizes[3] / n;        // 4

  // workspace layout (floats): PA[n] | PB[n] | Xt[n*batch] | TMP[rdim*n*batch]
  float* PA  = (float*)d_ws;
  float* PB  = PA + n;
  float* Xt  = PB + n;
  float* TMP = Xt + (size_t)n * batch;

  const size_t ldsBytes = (size_t)(PAD + n + 4) * sizeof(float);

  k_prefix<<<1, 64, 0, stream>>>(sA, sB, PA, PB, n);

  {
    dim3 grid(n / 32, batch / 32), block(32, 8);
    k_transpose_scale<<<grid, block, 0, stream>>>(x, PB, Xt, n, batch);
  }
  {
    dim3 grid(n / 64, batch / 256, rdim);
    k_pass1<<<grid, 256, ldsBytes, stream>>>(H, PB, Xt, TMP, n, batch);
  }
  {
    dim3 grid(n / 64, batch / 256, 1);
    k_pass2<<<grid, 256, ldsBytes, stream>>>(G, PA, TMP, bias, out, n, batch, rdim);
  }
}